// Sampler_55164559950217
// MI455X (gfx1250) — compile-verified
//
#include <hip/hip_runtime.h>
#include <math.h>

// Problem constants (from reference)
#define NWAY   64
#define KSHOT  5
#define QSHOT  1024
#define TOPK   32
#define DIM    768
#define NSUP   (NWAY * KSHOT)        // 320
#define NQRY   (NWAY * QSHOT)        // 65536
#define CHUNK  4096                  // N columns per GEMM/topk chunk
#define NCHUNK (NQRY / CHUNK)        // 16
#define CANDS  (NCHUNK * TOPK)       // 512 candidates per row
#define NEG_INF (-3.0e38f)

// GEMM tiling
#define MBLK  64                     // A rows staged per block (4 M-subtiles/wave)
#define KC    192                    // K chunk staged in LDS (768 = 4 * 192)
#define LSTR  196                    // LDS row stride: 196 % 64 == 4 -> lanes hit
                                     // banks spaced by 4, conflict-free b64 reads

typedef __attribute__((ext_vector_type(2))) float v2f;
typedef __attribute__((ext_vector_type(8))) float v8f;

// ---------------------------------------------------------------------------
// Row L2-normalization. One block per row. Writes normalized row (if outRows)
// and/or the inverse norm (if outInv).
// ---------------------------------------------------------------------------
__global__ __launch_bounds__(256) void norm_rows(const float* __restrict__ in,
                                                 float* __restrict__ outRows,
                                                 float* __restrict__ outInv) {
  __shared__ float red[256];
  const int row = blockIdx.x;
  const int t   = threadIdx.x;
  const float* r = in + (size_t)row * DIM;

  float ss = 0.0f;
  for (int i = t; i < DIM; i += 256) {
    float v = r[i];
    ss += v * v;
  }
  red[t] = ss;
  __syncthreads();
  for (int s = 128; s > 0; s >>= 1) {
    if (t < s) red[t] += red[t + s];
    __syncthreads();
  }
  const float inv = 1.0f / fmaxf(sqrtf(red[0]), 1e-12f);

  if (outRows) {
    float* o = outRows + (size_t)row * DIM;
    for (int i = t; i < DIM; i += 256) o[i] = r[i] * inv;
  }
  if (outInv && t == 0) outInv[row] = inv;
}

// ---------------------------------------------------------------------------
// GEMM chunk via V_WMMA_F32_16X16X4_F32 with 4-way B register reuse.
//  C[m][n] = (snorm[m] . q[n]) * qinv[n]  for n in [nbase, nbase+CHUNK)
// Grid: (NSUP/MBLK = 5, CHUNK/64 = 64), block = 128 threads = 4 waves.
// Each wave owns one 16-wide N-tile and accumulates FOUR 16x16 C tiles
// (M-subtiles) against a single B fragment per k-step:
//    per k-step: 1 global b64 (B) + 4 ds b64 (A) + 4 wmma  -> 4x the
//    arithmetic intensity of the 1:1 version, independent accumulators.
// A staged in LDS in K-chunks of 192 (64 x 196 floats = 50176 B).
// A layout (32-bit, 16x4): lanes 0-15 = M rows, v0/v1 = K+0/K+1;
//                          lanes 16-31 = same M rows, K+2/K+3.
// ---------------------------------------------------------------------------
__global__ __launch_bounds__(128) void gemm_chunk(const float* __restrict__ snorm,
                                                  const float* __restrict__ q,
                                                  const float* __restrict__ qinv,
                                                  float* __restrict__ simchunk,
                                                  int nbase) {
  __shared__ float aT[MBLK * LSTR];

  const int mb  = blockIdx.x;      // 0..4  (M block of 64 rows)
  const int tid = threadIdx.x;     // 0..127

  const int wave  = tid >> 5;
  const int lane  = tid & 31;
  const int lhalf = lane >> 4;     // 0: K+0/K+1 ; 1: K+2/K+3
  const int l15   = lane & 15;

  const int n0 = nbase + (blockIdx.y * 4 + wave) * 16;
  const float* brow = q + (size_t)(n0 + l15) * DIM + 2 * lhalf;
  const float* ablk = snorm + (size_t)mb * MBLK * DIM;

  v8f acc0 = {}, acc1 = {}, acc2 = {}, acc3 = {};

  for (int kc = 0; kc < DIM; kc += KC) {
    // Cooperative load of the 64 x 192 A chunk into LDS (coalesced).
    for (int i = tid; i < MBLK * KC; i += 128) {
      const int r = i / KC;
      const int c = i - r * KC;
      aT[r * LSTR + c] = ablk[(size_t)r * DIM + kc + c];
    }
    __syncthreads();

    const float* apt = aT + l15 * LSTR + 2 * lhalf;
    for (int k = 0; k < KC; k += 4) {
      const v2f b = *(const v2f*)(brow + kc + k);  // one B fragment, reused 4x
      v2f a0 = *(const v2f*)(apt + (0 * 16) * LSTR + k);
      v2f a1 = *(const v2f*)(apt + (1 * 16) * LSTR + k);
      v2f a2 = *(const v2f*)(apt + (2 * 16) * LSTR + k);
      v2f a3 = *(const v2f*)(apt + (3 * 16) * LSTR + k);
      acc0 = __builtin_amdgcn_wmma_f32_16x16x4_f32(false, a0, false, b,
                                                   (short)0, acc0, false, false);
      acc1 = __builtin_amdgcn_wmma_f32_16x16x4_f32(false, a1, false, b,
                                                   (short)0, acc1, false, false);
      acc2 = __builtin_amdgcn_wmma_f32_16x16x4_f32(false, a2, false, b,
                                                   (short)0, acc2, false, false);
      acc3 = __builtin_amdgcn_wmma_f32_16x16x4_f32(false, a3, false, b,
                                                   (short)0, acc3, false, false);
    }
    __syncthreads();
  }

  // Epilogue: scale by 1/||q|| and store.
  // C layout: lane holds C[m = v + 8*lhalf][n = l15] in acc[v].
  const float qi   = qinv[n0 + l15];
  const int   ncol = (n0 - nbase) + l15;
  const int   mrow = mb * MBLK + 8 * lhalf;
  v8f accs[4] = {acc0, acc1, acc2, acc3};
#pragma unroll
  for (int ms = 0; ms < 4; ++ms) {
    float* outp = simchunk + (size_t)(mrow + ms * 16) * CHUNK + ncol;
#pragma unroll
    for (int v = 0; v < 8; ++v) {
      outp[(size_t)v * CHUNK] = accs[ms][v] * qi;
    }
  }
}

// ---------------------------------------------------------------------------
// Per-chunk top-32 per support row. One block per row (320 blocks).
// Destructive iterative argmax in LDS; exact lax.top_k ordering:
// value descending, ties broken by smaller column index.
// ---------------------------------------------------------------------------
__global__ __launch_bounds__(256) void topk_chunk(const float* __restrict__ simchunk,
                                                  float* __restrict__ candVal,
                                                  int* __restrict__ candIdx,
                                                  int nbase, int chunkSlot) {
  __shared__ float sv[CHUNK];
  __shared__ float rv[256];
  __shared__ int   ri[256];

  const int r = blockIdx.x;
  const int t = threadIdx.x;

  const float* src = simchunk + (size_t)r * CHUNK;
  for (int i = t; i < CHUNK; i += 256) sv[i] = src[i];
  __syncthreads();

  for (int round = 0; round < TOPK; ++round) {
    float bv = NEG_INF;
    int   bj = 0x7fffffff;
    for (int j = t; j < CHUNK; j += 256) {
      const float v = sv[j];
      if (v > bv || (v == bv && j < bj)) { bv = v; bj = j; }
    }
    rv[t] = bv; ri[t] = bj;
    __syncthreads();
    for (int s = 128; s > 0; s >>= 1) {
      if (t < s) {
        const float ov = rv[t + s];
        const int   oj = ri[t + s];
        if (ov > rv[t] || (ov == rv[t] && oj < ri[t])) { rv[t] = ov; ri[t] = oj; }
      }
      __syncthreads();
    }
    if (t == 0) {
      const int wj = ri[0];
      const size_t o = (size_t)r * CANDS + chunkSlot * TOPK + round;
      candVal[o] = rv[0];
      candIdx[o] = nbase + wj;
      sv[wj] = NEG_INF;   // mask winner
    }
    __syncthreads();
  }
}

// ---------------------------------------------------------------------------
// Final top-32 per row over the 512 chunk candidates + per-row hit count.
// ---------------------------------------------------------------------------
__global__ __launch_bounds__(256) void topk_final(const float* __restrict__ candVal,
                                                  const int* __restrict__ candIdx,
                                                  int* __restrict__ outIdx,
                                                  int* __restrict__ hits) {
  __shared__ float sv[CANDS];
  __shared__ int   si[CANDS];
  __shared__ float rv[256];
  __shared__ int   ri[256];
  __shared__ int   rl[256];

  const int r = blockIdx.x;
  const int t = threadIdx.x;

  for (int i = t; i < CANDS; i += 256) {
    sv[i] = candVal[(size_t)r * CANDS + i];
    si[i] = candIdx[(size_t)r * CANDS + i];
  }
  __syncthreads();

  const int way = r / KSHOT;
  const int lo  = way * QSHOT;
  int hcount = 0;

  for (int round = 0; round < TOPK; ++round) {
    float bv = NEG_INF;
    int   bi = 0x7fffffff;
    int   bl = 0;
    for (int j = t; j < CANDS; j += 256) {
      const float v = sv[j];
      const int   g = si[j];
      if (v > bv || (v == bv && g < bi)) { bv = v; bi = g; bl = j; }
    }
    rv[t] = bv; ri[t] = bi; rl[t] = bl;
    __syncthreads();
    for (int s = 128; s > 0; s >>= 1) {
      if (t < s) {
        if (rv[t + s] > rv[t] || (rv[t + s] == rv[t] && ri[t + s] < ri[t])) {
          rv[t] = rv[t + s]; ri[t] = ri[t + s]; rl[t] = rl[t + s];
        }
      }
      __syncthreads();
    }
    if (t == 0) {
      const int wi = ri[0];
      outIdx[r * TOPK + round] = wi;
      sv[rl[0]] = NEG_INF;
      if (wi >= lo && wi < lo + QSHOT) hcount++;
    }
    __syncthreads();
  }
  if (t == 0) hits[r] = hcount;
}

// ---------------------------------------------------------------------------
// Gather raw query rows by selected index (float4 copies) + write acc scalar.
// ---------------------------------------------------------------------------
__global__ __launch_bounds__(192) void gather_out(const float4* __restrict__ q4,
                                                  const int* __restrict__ idx,
                                                  const int* __restrict__ hits,
                                                  float* __restrict__ out) {
  const int j = blockIdx.x;     // 0..10239 output rows
  const int t = threadIdx.x;    // 0..191 float4s per row (768 floats)
  const int src = idx[j];
  float4* out4 = (float4*)out;
  out4[(size_t)j * (DIM / 4) + t] = q4[(size_t)src * (DIM / 4) + t];

  if (j == 0 && t == 0) {
    int s = 0;
    for (int i = 0; i < NSUP; ++i) s += hits[i];
    out[(size_t)NSUP * TOPK * DIM] = (float)s / (float)(NSUP * TOPK);
  }
}

// ---------------------------------------------------------------------------
// Host-side launch. Workspace layout (all offsets 256B aligned, total ~7.9MB):
//   snorm    : 320*768 f32      @ 0
//   qinv     : 65536 f32        @ 983040
//   simchunk : 320*4096 f32     @ 1245184
//   candVal  : 320*512 f32      @ 6488064
//   candIdx  : 320*512 i32      @ 7143424
//   outIdx   : 320*32  i32      @ 7798784
//   hits     : 320     i32      @ 7839744
// ---------------------------------------------------------------------------
extern "C" void kernel_launch(void* const* d_in, const int* in_sizes, int n_in,
                              void* d_out, int out_size, void* d_ws, size_t ws_size,
                              hipStream_t stream) {
  (void)in_sizes; (void)n_in; (void)out_size; (void)ws_size;

  const float* sup = (const float*)d_in[0];   // (320, 768) fp32
  const float* qry = (const float*)d_in[1];   // (65536, 768) fp32
  float* out = (float*)d_out;                 // 10240*768 + 1 fp32

  char* ws = (char*)d_ws;
  float* snorm    = (float*)(ws + 0);
  float* qinv     = (float*)(ws + 983040);
  float* simchunk = (float*)(ws + 1245184);
  float* candVal  = (float*)(ws + 6488064);
  int*   candIdx  = (int*)  (ws + 7143424);
  int*   outIdx   = (int*)  (ws + 7798784);
  int*   hits     = (int*)  (ws + 7839744);

  norm_rows<<<NSUP, 256, 0, stream>>>(sup, snorm, nullptr);
  norm_rows<<<NQRY, 256, 0, stream>>>(qry, nullptr, qinv);

  for (int c = 0; c < NCHUNK; ++c) {
    dim3 g(NSUP / MBLK, CHUNK / 64);    // (5, 64), 4 waves/block, 4 M-subtiles/wave
    gemm_chunk<<<g, 128, 0, stream>>>(snorm, qry, qinv, simchunk, c * CHUNK);
    topk_chunk<<<NSUP, 256, 0, stream>>>(simchunk, candVal, candIdx, c * CHUNK, c);
  }

  topk_final<<<NSUP, 256, 0, stream>>>(candVal, candIdx, outIdx, hits);
  gather_out<<<NSUP * TOPK, 192, 0, stream>>>((const float4*)qry, outIdx, hits, out);
}